// Semantic_Mapping_48954037240461
// MI455X (gfx1250) — compile-verified
//
#include <hip/hip_runtime.h>
#include <stdint.h>

// ---------------- problem constants ----------------
#define BB    8
#define HH    480
#define WW    640
#define NPIX  (HH*WW)        // 307200
#define NSEMV 16
#define CCH   21             // obs channels
#define MCH   20             // map channels (C-1)
#define MS    240            // MAP_SIZE
#define VRV   100
#define ZD    80
#define NPTS  4096
#define MINZ  13
#define MAXZ  25
#define AGH   88.0f
#define DEGV  57.29577951308232f
#define PIF   3.14159265358979323846f

// ---------------- output layout (floats) ----------------
#define OFF_FP    0            // (8,1,100,100)   = 80000
#define OFF_MAP   80000        // (8,20,240,240)  = 9216000
#define OFF_POSE1 9296000      // (8,3)
#define OFF_POSE2 9296024      // (8,3)
#define OFF_FMP   9296048      // (8,7,4096)      = 229376
#define OFF_OMP   9525424      // (8,7,4096)

// ---------------- workspace layout (floats) ----------------
#define WS_PA    0             // proj_agent (8,17,100,100) = 1360000
#define WS_PL    1360000       // proj_all
#define WS_AV    2720000       // agent_view (8,20,240,240) = 9216000
#define WS_ROT   11936000      // rotated
#define WS_KEYS  21152000      // (8,307200)
#define WS_POSE  23609600      // (8,3)
#define WS_PAR   23609632      // (8,4): cos,sin,shx,shy
#define WS_SEL   23609664      // int (8,4096)

typedef unsigned int u32x4 __attribute__((ext_vector_type(4)));
typedef int          i32x8 __attribute__((ext_vector_type(8)));
typedef int          i32x4 __attribute__((ext_vector_type(4)));

__device__ __forceinline__ float dev_fcam() {
  return (WW * 0.5f) / tanf(0.5f * 79.0f * PIF / 180.0f);
}

__device__ __forceinline__ void pc_from_pixel(int h, int w, float depth,
                                              float& X, float& Y, float& Z) {
  float inv = depth / dev_fcam();
  X = ((float)w - (WW - 1.0f) * 0.5f) * inv;
  Y = depth;
  Z = ((float)(HH - 1 - h) - (HH - 1.0f) * 0.5f) * inv;
}

// ---------------- kernel 0: zero projection maps ----------------
__global__ __launch_bounds__(256) void k_zero(float* __restrict__ p, int n) {
  int i = blockIdx.x * 256 + threadIdx.x;
  if (i < n) p[i] = 0.0f;
}

// ---------------- kernel 1: fused splat + z-projection ----------------
__global__ __launch_bounds__(256) void k_splat(const float* __restrict__ obs,
                                               float* __restrict__ pa,
                                               float* __restrict__ pl) {
  int i = blockIdx.x * 256 + threadIdx.x;
  if (i >= BB * NPIX) return;
  int b = i / NPIX, p = i - b * NPIX;
  int h = p / WW, w = p - h * WW;
  const float* ob = obs + (size_t)b * CCH * NPIX;

  // stream-ahead prefetch of the depth channel (global_prefetch_b8)
  __builtin_prefetch(ob + 3 * NPIX + p + 4096, 0, 0);

  float X, Y, Z;
  pc_from_pixel(h, w, ob[3 * NPIX + p], X, Y, Z);
  float p0 = (X + 250.0f) * 0.2f;          // x voxel pos
  float p1 = Y * 0.2f;                     // y voxel pos
  float p2 = (Z + AGH) * 0.2f + 8.0f;      // z voxel pos
  float f0 = floorf(p0), f1 = floorf(p1), f2 = floorf(p2);
  int ix = (int)f0, iy = (int)f1, iz = (int)f2;
  float fx = p0 - f0, fy = p1 - f1, fz = p2 - f2;

  // combine the two z-corners into (all, agent-window) weights
  float wzA = 0.0f, wzG = 0.0f;
  #pragma unroll
  for (int d = 0; d < 2; ++d) {
    int pz = iz + d;
    if (pz > 0 && pz < ZD) {
      float wz = d ? fz : (1.0f - fz);
      wzA += wz;
      if (pz >= MINZ && pz < MAXZ) wzG += wz;
    }
  }
  if (wzA == 0.0f) return;

  float feat[17];
  feat[0] = 1.0f;
  #pragma unroll
  for (int s = 0; s < NSEMV; ++s) feat[1 + s] = ob[(4 + s) * NPIX + p];

  #pragma unroll
  for (int dy = 0; dy < 2; ++dy) {
    int py = iy + dy;
    if (py <= 0 || py >= VRV) continue;
    float wy = dy ? fy : (1.0f - fy);
    #pragma unroll
    for (int dx = 0; dx < 2; ++dx) {
      int px = ix + dx;
      if (px <= 0 || px >= VRV) continue;
      float wxy = (dx ? fx : (1.0f - fx)) * wy;
      if (wxy == 0.0f) continue;
      float wa = wxy * wzA, wg = wxy * wzG;
      size_t cell = (size_t)b * 170000 + (size_t)py * 100 + px;
      #pragma unroll
      for (int f = 0; f < 17; ++f) {
        float v = feat[f];
        atomicAdd(&pl[cell + (size_t)f * 10000], v * wa);
        if (wg != 0.0f) atomicAdd(&pa[cell + (size_t)f * 10000], v * wg);
      }
    }
  }
}

// ---------------- kernel 2: pose update + sampling params ----------------
__global__ __launch_bounds__(32) void k_pose(const float* __restrict__ pose_obs,
                                             const float* __restrict__ poses_last,
                                             float* __restrict__ poses_ws,
                                             float* __restrict__ params,
                                             float* __restrict__ out) {
  int b = threadIdx.x;
  if (b >= BB) return;
  float ox = pose_obs[b * 3 + 0], oy = pose_obs[b * 3 + 1], ot = pose_obs[b * 3 + 2];
  float lx = poses_last[b * 3 + 0], ly = poses_last[b * 3 + 1], lt = poses_last[b * 3 + 2];
  float rad = lt / DEGV;
  float s = sinf(rad), c = cosf(rad);
  float ny = ly + ox * s + oy * c;
  float nx = lx + ox * c - oy * s;
  float nt = lt + ot * DEGV;
  nt = fmodf(nt - 180.0f, 360.0f) + 180.0f;
  nt = fmodf(nt + 180.0f, 360.0f) - 180.0f;
  poses_ws[b * 3 + 0] = nx; poses_ws[b * 3 + 1] = ny; poses_ws[b * 3 + 2] = nt;
  out[OFF_POSE1 + b * 3 + 0] = nx; out[OFF_POSE1 + b * 3 + 1] = ny; out[OFF_POSE1 + b * 3 + 2] = nt;
  out[OFF_POSE2 + b * 3 + 0] = nx; out[OFF_POSE2 + b * 3 + 1] = ny; out[OFF_POSE2 + b * 3 + 2] = nt;
  float stt = (90.0f - nt) * PIF / 180.0f;
  params[b * 4 + 0] = cosf(stt);
  params[b * 4 + 1] = sinf(stt);
  float tx = -(nx * 100.0f / 5.0f - 120.0f) / 120.0f;  // st_xy.x
  float ty = -(ny * 100.0f / 5.0f - 120.0f) / 120.0f;  // st_xy.y
  params[b * 4 + 2] = tx * 119.5f;   // pixel shift x
  params[b * 4 + 3] = ty * 119.5f;   // pixel shift y
}

// ---------------- kernel 3: build agent_view ----------------
__global__ __launch_bounds__(256) void k_agent_view(const float* __restrict__ pa,
                                                    const float* __restrict__ pl,
                                                    float* __restrict__ av) {
  int i = blockIdx.x * 256 + threadIdx.x;
  if (i >= BB * MCH * MS * MS) return;
  int x = i % MS, t = i / MS;
  int y = t % MS; t /= MS;
  int c = t % MCH, b = t / MCH;
  float v = 0.0f;
  if (y >= 120 && y < 220 && x >= 70 && x < 170) {
    int yy = y - 120, xx = x - 70;
    size_t cb = (size_t)b * 170000 + (size_t)yy * 100 + xx;
    if (c == 0)      v = fminf(fmaxf(pa[cb], 0.0f), 1.0f);
    else if (c == 1) v = fminf(fmaxf(pl[cb], 0.0f), 1.0f);
    else if (c >= 4) v = fminf(fmaxf(pa[cb + (size_t)(c - 3) * 10000] * 0.2f, 0.0f), 1.0f);
  }
  av[i] = v;
}

// ---------------- kernel 4: fp_map_pred output ----------------
__global__ __launch_bounds__(256) void k_fp(const float* __restrict__ pa, float* __restrict__ out) {
  int i = blockIdx.x * 256 + threadIdx.x;
  if (i >= BB * 10000) return;
  int b = i / 10000, r = i - b * 10000;
  out[OFF_FP + i] = fminf(fmaxf(pa[(size_t)b * 170000 + r], 0.0f), 1.0f);
}

// ---------------- kernel 5: rotation grid_sample ----------------
__global__ __launch_bounds__(256) void k_rot(const float* __restrict__ av,
                                             const float* __restrict__ params,
                                             float* __restrict__ rot) {
  int i = blockIdx.x * 256 + threadIdx.x;
  if (i >= BB * MCH * MS * MS) return;
  int x = i % MS, t = i / MS;
  int y = t % MS; t /= MS;
  int c = t % MCH, b = t / MCH;
  float ct = params[b * 4 + 0], st = params[b * 4 + 1];
  float gx = (float)x * (2.0f / 239.0f) - 1.0f;
  float gy = (float)y * (2.0f / 239.0f) - 1.0f;
  float xs = (ct * gx - st * gy + 1.0f) * 119.5f;
  float ys = (st * gx + ct * gy + 1.0f) * 119.5f;
  float xf = floorf(xs), yf = floorf(ys);
  const float* img = av + ((size_t)b * MCH + c) * MS * MS;
  float acc = 0.0f;
  #pragma unroll
  for (int dy = 0; dy < 2; ++dy)
    #pragma unroll
    for (int dx = 0; dx < 2; ++dx) {
      float xi = xf + dx, yi = yf + dy;
      if (xi >= 0.0f && xi <= 239.0f && yi >= 0.0f && yi <= 239.0f) {
        float wgt = (1.0f - fabsf(xs - xi)) * (1.0f - fabsf(ys - yi));
        acc += wgt * img[(int)yi * MS + (int)xi];
      }
    }
  rot[i] = acc;
}

// ---------------- kernel 6: translate + max ----------------
// Input band staged to LDS. Fast path: one TDM descriptor (tensor_load_to_lds,
// TENSORcnt) moves the whole 18x240 band; the D# OOB rule zero-fills rows past
// the image bottom. Edge path (band starts above/below the image): per-lane
// global_load_async_to_lds_b128 (ASYNCcnt) with manual zero-fill.
__global__ __launch_bounds__(256) void k_trans(const float* __restrict__ rot,
                                               const float* __restrict__ maps_last,
                                               const float* __restrict__ params,
                                               float* __restrict__ outmap) {
  // block -> (b, c, band of 16 output rows)
  int blk = blockIdx.x;
  int band = blk % 15; int t = blk / 15;
  int c = t % MCH, b = t / MCH;
  float shx = params[b * 4 + 2], shy = params[b * 4 + 3];
  int y0 = band * 16;
  int rlo = (int)floorf((float)y0 + shy);
  int rlo_u = __builtin_amdgcn_readfirstlane(rlo);   // uniform for descriptor build

  alignas(16) __shared__ float tile[18 * MS];   // 18 rows x 240 cols = 17.3 KB
  const float* plane = rot + ((size_t)b * MCH + c) * MS * MS;
  unsigned ldsbase = (unsigned)(uintptr_t)(&tile[0]);   // low 32 bits == LDS offset

  if (rlo_u >= 0 && rlo_u < MS) {
    // ---- TDM fast path: single tensor DMA for the band ----
    if (threadIdx.x == 0) {
      unsigned long long ga =
          (unsigned long long)(const void*)(plane + (size_t)rlo_u * MS);
      int nrows = MS - rlo_u;            // tensor rows available below rlo
      u32x4 g0;
      g0.x = 1u;                                   // count=1 (valid user D#)
      g0.y = ldsbase;                              // lds_addr (bytes)
      g0.z = (unsigned)(ga & 0xffffffffu);         // global_addr[31:0]
      g0.w = (unsigned)((ga >> 32) & 0x01ffffffu)  // global_addr[56:32]
             | 0x80000000u;                        // type=2 ("image")
      i32x8 g1;
      g1[0] = (2 << 16);                  // data_size=4B; wg_mask=0; no flags
      g1[1] = (MS << 16);                 // tensor_dim0 = 240 (low 16)
      g1[2] = ((nrows & 0xffff) << 16);   // tensor_dim0 hi=0 | tensor_dim1 lo
      g1[3] = (MS << 16);                 // tensor_dim1 hi=0 | tile_dim0 = 240
      g1[4] = 18;                         // tile_dim1 = 18, tile_dim2 = 0
      g1[5] = MS;                         // tensor_dim0_stride = 240 (low 32)
      g1[6] = 0;                          // stride hi, tensor_dim1_stride lo
      g1[7] = 0;
      i32x4 g2 = {0, 0, 0, 0};            // dims 2/3 unused (tile_dim3 = 0)
      i32x4 g3 = {0, 0, 0, 0};
      i32x8 g4 = {0, 0, 0, 0, 0, 0, 0, 0};
      __builtin_amdgcn_tensor_load_to_lds(g0, g1, g2, g3, g4, 0);
      __builtin_amdgcn_s_wait_tensorcnt(0);
    }
  } else {
    // ---- edge path: per-lane async copies + zero fill ----
    for (int ch = (int)threadIdx.x; ch < 18 * 60; ch += 256) {
      int k = ch / 60;
      int c4 = (ch - k * 60) * 4;
      int rr = rlo + k;
      if (rr >= 0 && rr < MS) {
        unsigned laddr = ldsbase + (unsigned)((k * MS + c4) * 4);
        unsigned long long g = (unsigned long long)(const void*)(plane + rr * MS + c4);
        asm volatile("global_load_async_to_lds_b128 %0, %1, off"
                     :: "v"(laddr), "v"(g) : "memory");
      } else {
        tile[k * MS + c4 + 0] = 0.0f; tile[k * MS + c4 + 1] = 0.0f;
        tile[k * MS + c4 + 2] = 0.0f; tile[k * MS + c4 + 3] = 0.0f;
      }
    }
    asm volatile("s_wait_asynccnt 0" ::: "memory");
  }
  __syncthreads();

  const float* ml = maps_last + ((size_t)b * MCH + c) * MS * MS;
  float* om = outmap + ((size_t)b * MCH + c) * MS * MS;
  for (int idx = (int)threadIdx.x; idx < 16 * MS; idx += 256) {
    int oy = y0 + idx / MS, ox = idx % MS;
    float xs = (float)ox + shx;
    float ys = (float)oy + shy;
    float xf = floorf(xs), yf = floorf(ys);
    float acc = 0.0f;
    #pragma unroll
    for (int dy = 0; dy < 2; ++dy)
      #pragma unroll
      for (int dx = 0; dx < 2; ++dx) {
        float xi = xf + dx, yi = yf + dy;
        if (xi >= 0.0f && xi <= 239.0f && yi >= 0.0f && yi <= 239.0f) {
          float wgt = (1.0f - fabsf(xs - xi)) * (1.0f - fabsf(ys - yi));
          acc += wgt * tile[((int)yi - rlo) * MS + (int)xi];
        }
      }
    om[oy * MS + ox] = fmaxf(ml[oy * MS + ox], acc);
  }
}

// ---------------- kernel 7: selection keys (log entropy + hash-gumbel) ----------------
__global__ __launch_bounds__(256) void k_keys(const float* __restrict__ obs,
                                              float* __restrict__ keys) {
  int i = blockIdx.x * 256 + threadIdx.x;
  if (i >= BB * NPIX) return;
  int b = i / NPIX, p = i - b * NPIX;
  const float* ob = obs + (size_t)b * CCH * NPIX;
  __builtin_prefetch(ob + 20 * NPIX + p + 4096, 0, 0);
  float depth = ob[3 * NPIX + p];
  float ent   = ob[20 * NPIX + p];
  float k;
  if (depth > 0.0f) {
    unsigned hsh = (unsigned)i * 0x9E3779B9u + 0x7F4A7C15u;
    hsh ^= hsh >> 16; hsh *= 0x85EBCA6Bu; hsh ^= hsh >> 13;
    hsh *= 0xC2B2AE35u; hsh ^= hsh >> 16;
    float u = ((float)(hsh >> 8) + 0.5f) * (1.0f / 16777216.0f);
    float g = -logf(-logf(u));
    k = logf(fmaxf(ent, 1e-30f)) + g;
  } else {
    k = -3.0e38f;
  }
  keys[i] = k;
}

// ---------------- kernel 8: deterministic top-NPTS per batch ----------------
__device__ __forceinline__ unsigned ordkey(float f) {
  unsigned u = __float_as_uint(f);
  return (u & 0x80000000u) ? ~u : (u | 0x80000000u);   // larger float -> larger uint
}

__global__ __launch_bounds__(1024) void k_select(const float* __restrict__ keys,
                                                 int* __restrict__ sel) {
  const int NT = 1024, CH = NPIX / NT;   // 300 contiguous elements / thread
  int b = blockIdx.x;
  const float* kb = keys + (size_t)b * NPIX;
  __shared__ unsigned hist[256];
  __shared__ unsigned scan[1024];
  __shared__ unsigned s_prefix, s_need, s_defA;
  if (threadIdx.x == 0) { s_prefix = 0u; s_need = NPTS; }
  __syncthreads();
  int c0 = (int)threadIdx.x * CH;

  // 3 rounds x 8 bits threshold refinement
  for (int round = 0; round < 3; ++round) {
    for (int q = (int)threadIdx.x; q < 256; q += NT) hist[q] = 0u;
    __syncthreads();
    unsigned pf = s_prefix;
    int shift = 24 - 8 * round;
    for (int j = c0; j < c0 + CH; ++j) {
      unsigned u = ordkey(kb[j]);
      unsigned hi = (round == 0) ? 0u : (u >> (shift + 8));
      if (hi == pf) atomicAdd(&hist[(u >> shift) & 255u], 1u);
    }
    __syncthreads();
    if (threadIdx.x == 0) {
      unsigned acc = 0, need = s_need;
      for (int q = 255; q >= 0; --q) {
        unsigned nv = acc + hist[q];
        if (nv >= need) { s_need = need - acc; s_prefix = (pf << 8) | (unsigned)q; break; }
        acc = nv;
      }
    }
    __syncthreads();
  }
  unsigned pf24 = s_prefix;
  int* sb = sel + (size_t)b * NPTS;

  // deterministic compact: strictly-above items, ordered by element index
  unsigned cntD = 0;
  for (int j = c0; j < c0 + CH; ++j) if ((ordkey(kb[j]) >> 8) > pf24) cntD++;
  scan[threadIdx.x] = cntD; __syncthreads();
  for (int off = 1; off < NT; off <<= 1) {
    unsigned v = (threadIdx.x >= (unsigned)off) ? scan[threadIdx.x - off] : 0u;
    __syncthreads();
    scan[threadIdx.x] += v;
    __syncthreads();
  }
  unsigned exclD = scan[threadIdx.x] - cntD;
  if (threadIdx.x == NT - 1) s_defA = scan[NT - 1];
  __syncthreads();
  unsigned A = s_defA;
  {
    unsigned pos = exclD;
    for (int j = c0; j < c0 + CH; ++j)
      if ((ordkey(kb[j]) >> 8) > pf24) { if (pos < NPTS) sb[pos] = j; pos++; }
  }
  __syncthreads();

  // fill remainder from threshold-equal items, ordered by element index
  unsigned cntE = 0;
  for (int j = c0; j < c0 + CH; ++j) if ((ordkey(kb[j]) >> 8) == pf24) cntE++;
  scan[threadIdx.x] = cntE; __syncthreads();
  for (int off = 1; off < NT; off <<= 1) {
    unsigned v = (threadIdx.x >= (unsigned)off) ? scan[threadIdx.x - off] : 0u;
    __syncthreads();
    scan[threadIdx.x] += v;
    __syncthreads();
  }
  unsigned exclE = scan[threadIdx.x] - cntE;
  {
    unsigned pos = A + exclE;
    for (int j = c0; j < c0 + CH; ++j)
      if ((ordkey(kb[j]) >> 8) == pf24) { if (pos < NPTS) sb[pos] = j; pos++; }
  }
}

// ---------------- kernel 9: gather selected points -> new_fmp / obs_map_points ----------------
__global__ __launch_bounds__(256) void k_fmp(const float* __restrict__ obs,
                                             const int* __restrict__ sel,
                                             const float* __restrict__ poses,
                                             float* __restrict__ out) {
  int i = blockIdx.x * 256 + threadIdx.x;
  if (i >= BB * NPTS) return;
  int b = i / NPTS, j = i - b * NPTS;
  int p = sel[i];
  int h = p / WW, w = p - h * WW;
  const float* ob = obs + (size_t)b * CCH * NPIX;
  float X, Y, Z;
  pc_from_pixel(h, w, ob[3 * NPIX + p], X, Y, Z);
  float nx = poses[b * 3 + 0], ny = poses[b * 3 + 1], nt = poses[b * 3 + 2];
  float ang = nt * PIF / 180.0f - 0.5f * PIF;
  float ca = cosf(ang), sa = sinf(ang);
  float wx = (X * ca - Y * sa + nx * 100.0f) * 0.2f;
  float wy = (X * sa + Y * ca + ny * 100.0f) * 0.2f;
  float wz = Z * 0.2f;
  float r3 = ob[0 * NPIX + p], r4 = ob[1 * NPIX + p], r5 = ob[2 * NPIX + p];
  float r6 = ob[20 * NPIX + p];

  float* nf = out + OFF_FMP + (size_t)b * 7 * NPTS;
  nf[0 * NPTS + j] = wx; nf[1 * NPTS + j] = wy; nf[2 * NPTS + j] = wz;
  nf[3 * NPTS + j] = r3; nf[4 * NPTS + j] = r4; nf[5 * NPTS + j] = r5;
  nf[6 * NPTS + j] = r6;

  float* op = out + OFF_OMP + (size_t)b * 7 * NPTS;
  op[0 * NPTS + j] = wx - nx * 100.0f * 0.2f;
  op[1 * NPTS + j] = wy - ny * 100.0f * 0.2f;
  op[2 * NPTS + j] = wz - AGH * 0.2f;
  op[3 * NPTS + j] = r3; op[4 * NPTS + j] = r4; op[5 * NPTS + j] = r5;
  op[6 * NPTS + j] = r6;
}

// ---------------- launcher ----------------
extern "C" void kernel_launch(void* const* d_in, const int* in_sizes, int n_in,
                              void* d_out, int out_size, void* d_ws, size_t ws_size,
                              hipStream_t stream) {
  (void)in_sizes; (void)n_in; (void)out_size; (void)ws_size;
  const float* obs        = (const float*)d_in[0];
  const float* pose_obs   = (const float*)d_in[1];
  const float* maps_last  = (const float*)d_in[2];
  const float* poses_last = (const float*)d_in[3];
  // d_in[4] origins, d_in[5] full_map_points (zeros), d_in[6] goal_cat_id: unused

  float* out = (float*)d_out;
  float* ws  = (float*)d_ws;
  float* pa     = ws + WS_PA;
  float* pl     = ws + WS_PL;
  float* av     = ws + WS_AV;
  float* rot    = ws + WS_ROT;
  float* keys   = ws + WS_KEYS;
  float* poses  = ws + WS_POSE;
  float* params = ws + WS_PAR;
  int*   sel    = (int*)(ws + WS_SEL);

  k_zero      <<<(2720000 + 255) / 256, 256, 0, stream>>>(pa, 2720000);
  k_pose      <<<1, 32, 0, stream>>>(pose_obs, poses_last, poses, params, out);
  k_splat     <<<(BB * NPIX) / 256, 256, 0, stream>>>(obs, pa, pl);
  k_agent_view<<<(BB * MCH * MS * MS) / 256, 256, 0, stream>>>(pa, pl, av);
  k_fp        <<<(BB * 10000 + 255) / 256, 256, 0, stream>>>(pa, out);
  k_rot       <<<(BB * MCH * MS * MS) / 256, 256, 0, stream>>>(av, params, rot);
  k_trans     <<<BB * MCH * 15, 256, 0, stream>>>(rot, maps_last, params, out + OFF_MAP);
  k_keys      <<<(BB * NPIX) / 256, 256, 0, stream>>>(obs, keys);
  k_select    <<<BB, 1024, 0, stream>>>(keys, sel);
  k_fmp       <<<(BB * NPTS) / 256, 256, 0, stream>>>(obs, sel, poses, out);
}